// ManifoldSECD_7842610282507
// MI455X (gfx1250) — compile-verified
//
#include <hip/hip_runtime.h>
#include <math.h>

// ---------------------------------------------------------------------------
// ManifoldSECD forward for MI455X (gfx1250, wave32, WMMA f32 16x16x4)
// B=256, D=1024, K=7 experts, VOC=32000
// ---------------------------------------------------------------------------

typedef float v2f __attribute__((ext_vector_type(2)));
typedef float v8f __attribute__((ext_vector_type(8)));

#define B_SZ 256
#define D_SZ 1024
#define KEXP 7
#define VOC_SZ 32000
#define MAXT 32   // max 16-row expert tiles (<= 256/16 + 6 padding tiles)

__device__ __forceinline__ v8f wmma_f32x4(v2f a, v2f b, v8f c) {
  // D = A(16x4 f32) x B(4x16 f32) + C(16x16 f32)
  return __builtin_amdgcn_wmma_f32_16x16x4_f32(false, a, false, b, (short)0, c,
                                               false, false);
}

// ---------------------------------------------------------------------------
// Workspace layout (float elements)
// ---------------------------------------------------------------------------
#define OFF_FEMB   ((size_t)0)          // [256*1024] f_emb
#define OFF_C      ((size_t)262144)     // [256*1024] c = h@U[k] + f@V[k]
#define OFF_ZA     ((size_t)524288)     // [256*1024] z ping
#define OFF_ZB     ((size_t)786432)     // [256*1024] z pong
#define OFF_AL     ((size_t)1048576)    // [256*1024] alpha_local
#define OFF_SH     ((size_t)1310720)    // [256*1024] stab_h
#define OFF_GAMMA  ((size_t)1572864)    // [256] gamma
#define OFF_EXPERT ((size_t)1573120)    // [256] int expert id
#define OFF_TEXP   ((size_t)1573376)    // [32] int tile expert
#define OFF_TROWS  ((size_t)1573408)    // [32*16] int tile row indices (-1 pad)
#define OFF_NT     ((size_t)1573920)    // [1] int tile count

// ---------------------------------------------------------------------------
// 1) Per-row prep: token/fiber embeddings, cosine routing, softmax pi,
//    argmax expert, entropy -> gamma MLP.  One block (256 thr) per row.
// ---------------------------------------------------------------------------
__global__ void prep_kernel(const float* __restrict__ h,
                            const float* __restrict__ fib_s,
                            const float* __restrict__ fib_e,
                            const int*   __restrict__ tok,
                            const float* __restrict__ op_emb,
                            const float* __restrict__ num_w,
                            const float* __restrict__ num_b,
                            const float* __restrict__ addr,
                            const float* __restrict__ fs_w,
                            const float* __restrict__ fs_b,
                            const float* __restrict__ fe_w,
                            const float* __restrict__ fe_b,
                            const float* __restrict__ c1w,
                            const float* __restrict__ c1b,
                            const float* __restrict__ c2w,
                            const float* __restrict__ c2b,
                            float* __restrict__ femb_out,
                            float* __restrict__ gamma_out,
                            int*   __restrict__ expert_out,
                            float* __restrict__ pi_out) {
  int b = blockIdx.x;
  int tid = threadIdx.x;
  int t = tok[b];
  bool isnum = (t >= 7);
  float nv = (float)t - 7.0f;
  int opk = t < 0 ? 0 : (t > 6 ? 6 : t);
  float s = fib_s[b], e = fib_e[b];

  // partial sums: 0:|te|^2 1:|te+f|^2 2..8:te.addr[k] 9..15:(te+f).addr[k]
  // 16..22:|addr[k]|^2
  float part[23];
#pragma unroll
  for (int q = 0; q < 23; ++q) part[q] = 0.f;

  for (int d = tid; d < D_SZ; d += 256) {
    float te = isnum ? (nv * num_w[d] + num_b[d]) : op_emb[opk * D_SZ + d];
    float fm = tanhf(s * fs_w[d] + fs_b[d] + e * fe_w[d] + fe_b[d]);
    femb_out[(size_t)b * D_SZ + d] = fm;
    float tf = te + fm;
    part[0] += te * te;
    part[1] += tf * tf;
#pragma unroll
    for (int k = 0; k < KEXP; ++k) {
      float a = addr[k * D_SZ + d];
      part[2 + k] += te * a;
      part[9 + k] += tf * a;
      part[16 + k] += a * a;
    }
  }

  __shared__ float wsum[8][23];
  __shared__ float red[23];
  int lane = tid & 31, wave = tid >> 5;
#pragma unroll
  for (int q = 0; q < 23; ++q) {
    float v = part[q];
#pragma unroll
    for (int off = 16; off > 0; off >>= 1) v += __shfl_xor(v, off, 32);
    if (lane == 0) wsum[wave][q] = v;
  }
  __syncthreads();
  if (tid < 23) {
    float sm = 0.f;
#pragma unroll
    for (int w = 0; w < 8; ++w) sm += wsum[w][tid];
    red[tid] = sm;
  }
  __syncthreads();

  if (tid == 0) {
    float nte = fmaxf(sqrtf(red[0]), 1e-12f);
    float ntf = fmaxf(sqrtf(red[1]), 1e-12f);
    bool use_ctrl = (t < 7);
    float sc[KEXP];
#pragma unroll
    for (int k = 0; k < KEXP; ++k) {
      float na = fmaxf(sqrtf(red[16 + k]), 1e-12f);
      sc[k] = use_ctrl ? red[2 + k] / (nte * na) : red[9 + k] / (ntf * na);
    }
    float mx = -1e30f;
#pragma unroll
    for (int k = 0; k < KEXP; ++k) mx = fmaxf(mx, 5.0f * sc[k]);
    float pi[KEXP], ssum = 0.f;
#pragma unroll
    for (int k = 0; k < KEXP; ++k) { pi[k] = expf(5.0f * sc[k] - mx); ssum += pi[k]; }
    float H = 0.f, best = -1.f;
    int am = 0;
#pragma unroll
    for (int k = 0; k < KEXP; ++k) {
      pi[k] /= ssum;
      pi_out[(size_t)b * KEXP + k] = pi[k];
      if (pi[k] > best) { best = pi[k]; am = k; }  // first-max like jnp.argmax
      H -= pi[k] * logf(pi[k] + 1e-8f);
    }
    expert_out[b] = am;
    // ctrl MLP: input [entropy, 0] -> 16 relu -> softplus scalar
    float g = c2b[0];
#pragma unroll
    for (int j = 0; j < 16; ++j) {
      float y = c1w[j * 2 + 0] * H + c1b[j];  // seq_position term is 0
      y = fmaxf(y, 0.f);
      g += c2w[j] * y;
    }
    gamma_out[b] = (g > 20.f) ? g : log1pf(expf(g));
  }
}

// ---------------------------------------------------------------------------
// 2) Group rows by expert into padded 16-row tiles (serial; B=256, trivial)
// ---------------------------------------------------------------------------
__global__ void group_kernel(const int* __restrict__ expert,
                             int* __restrict__ tile_expert,
                             int* __restrict__ tile_rows,
                             int* __restrict__ n_tiles) {
  if (threadIdx.x != 0 || blockIdx.x != 0) return;
  int cnt[KEXP], off[KEXP], pos[KEXP];
  for (int k = 0; k < KEXP; ++k) cnt[k] = 0;
  for (int b = 0; b < B_SZ; ++b) cnt[expert[b]]++;
  int acc = 0;
  for (int k = 0; k < KEXP; ++k) { off[k] = acc; acc += cnt[k]; pos[k] = off[k]; }
  int perm[B_SZ];
  for (int b = 0; b < B_SZ; ++b) perm[pos[expert[b]]++] = b;  // stable by row
  int tc = 0;
  for (int k = 0; k < KEXP; ++k)
    for (int s = 0; s < cnt[k]; s += 16) {
      tile_expert[tc] = k;
      for (int j = 0; j < 16; ++j)
        tile_rows[tc * 16 + j] = (s + j < cnt[k]) ? perm[off[k] + s + j] : -1;
      ++tc;
    }
  *n_tiles = tc;
}

// ---------------------------------------------------------------------------
// 3) c = h @ U[k_b] + f_emb @ V[k_b];  z1 = tanh(c)
//    grid (MAXT, 4), block (32,4): wave handles 4 col-tiles (64 cols)
// ---------------------------------------------------------------------------
__global__ void deq_const_kernel(const float* __restrict__ hbuf,
                                 const float* __restrict__ femb,
                                 const float* __restrict__ Ubase,
                                 const float* __restrict__ Vbase,
                                 const int* __restrict__ tile_expert,
                                 const int* __restrict__ tile_rows,
                                 const int* __restrict__ n_tiles,
                                 float* __restrict__ cbuf,
                                 float* __restrict__ z1) {
  int tile = blockIdx.x;
  if (tile >= *n_tiles) return;
  __shared__ int rws[16];
  int lin = threadIdx.y * 32 + threadIdx.x;
  if (lin < 16) rws[lin] = tile_rows[tile * 16 + lin];
  __syncthreads();
  int kexp = tile_expert[tile];
  const float* U = Ubase + (size_t)kexp * D_SZ * D_SZ;
  const float* V = Vbase + (size_t)kexp * D_SZ * D_SZ;
  int lane = threadIdx.x;
  int m = lane & 15, kh = lane >> 4;
  int colbase = blockIdx.y * 256 + threadIdx.y * 64;
  int ar = rws[m]; if (ar < 0) ar = 0;

  v8f acc[4];
#pragma unroll
  for (int ct = 0; ct < 4; ++ct)
#pragma unroll
    for (int i = 0; i < 8; ++i) acc[ct][i] = 0.f;

  for (int kb = 0; kb < D_SZ; kb += 4) {
    int kk = kb + 2 * kh;
    v2f a;
    a.x = hbuf[(size_t)ar * D_SZ + kk];
    a.y = hbuf[(size_t)ar * D_SZ + kk + 1];
#pragma unroll
    for (int ct = 0; ct < 4; ++ct) {
      int n = colbase + ct * 16 + m;
      v2f bv;
      bv.x = U[(size_t)kk * D_SZ + n];
      bv.y = U[(size_t)(kk + 1) * D_SZ + n];
      acc[ct] = wmma_f32x4(a, bv, acc[ct]);
    }
  }
  for (int kb = 0; kb < D_SZ; kb += 4) {
    int kk = kb + 2 * kh;
    v2f a;
    a.x = femb[(size_t)ar * D_SZ + kk];
    a.y = femb[(size_t)ar * D_SZ + kk + 1];
#pragma unroll
    for (int ct = 0; ct < 4; ++ct) {
      int n = colbase + ct * 16 + m;
      v2f bv;
      bv.x = V[(size_t)kk * D_SZ + n];
      bv.y = V[(size_t)(kk + 1) * D_SZ + n];
      acc[ct] = wmma_f32x4(a, bv, acc[ct]);
    }
  }
#pragma unroll
  for (int ct = 0; ct < 4; ++ct) {
    int n = colbase + ct * 16 + m;
#pragma unroll
    for (int i = 0; i < 8; ++i) {
      int r = rws[i + 8 * kh];
      if (r >= 0) {
        float v = acc[ct][i];
        cbuf[(size_t)r * D_SZ + n] = v;
        z1[(size_t)r * D_SZ + n] = tanhf(v);
      }
    }
  }
}

// ---------------------------------------------------------------------------
// 4) One DEQ iteration: z_out = tanh(z_in @ W[k_b] + c)
// ---------------------------------------------------------------------------
__global__ void deq_iter_kernel(const float* __restrict__ zin,
                                float* __restrict__ zout,
                                const float* __restrict__ cbuf,
                                const float* __restrict__ Wbase,
                                const int* __restrict__ tile_expert,
                                const int* __restrict__ tile_rows,
                                const int* __restrict__ n_tiles) {
  int tile = blockIdx.x;
  if (tile >= *n_tiles) return;
  __shared__ int rws[16];
  int lin = threadIdx.y * 32 + threadIdx.x;
  if (lin < 16) rws[lin] = tile_rows[tile * 16 + lin];
  __syncthreads();
  int kexp = tile_expert[tile];
  const float* Wk = Wbase + (size_t)kexp * D_SZ * D_SZ;
  int lane = threadIdx.x;
  int m = lane & 15, kh = lane >> 4;
  int colbase = blockIdx.y * 256 + threadIdx.y * 64;
  int ar = rws[m]; if (ar < 0) ar = 0;

  v8f acc[4];
#pragma unroll
  for (int ct = 0; ct < 4; ++ct) {
    int n = colbase + ct * 16 + m;
#pragma unroll
    for (int i = 0; i < 8; ++i) {
      int r = rws[i + 8 * kh]; if (r < 0) r = 0;
      acc[ct][i] = cbuf[(size_t)r * D_SZ + n];   // C preload = constant term
    }
  }
  for (int kb = 0; kb < D_SZ; kb += 4) {
    int kk = kb + 2 * kh;
    v2f a;
    a.x = zin[(size_t)ar * D_SZ + kk];
    a.y = zin[(size_t)ar * D_SZ + kk + 1];
#pragma unroll
    for (int ct = 0; ct < 4; ++ct) {
      int n = colbase + ct * 16 + m;
      v2f bv;
      bv.x = Wk[(size_t)kk * D_SZ + n];
      bv.y = Wk[(size_t)(kk + 1) * D_SZ + n];
      acc[ct] = wmma_f32x4(a, bv, acc[ct]);
    }
  }
#pragma unroll
  for (int ct = 0; ct < 4; ++ct) {
    int n = colbase + ct * 16 + m;
#pragma unroll
    for (int i = 0; i < 8; ++i) {
      int r = rws[i + 8 * kh];
      if (r >= 0) zout[(size_t)r * D_SZ + n] = tanhf(acc[ct][i]);
    }
  }
}

// ---------------------------------------------------------------------------
// 5) Generic y = act( [x1|x2] @ Wt^T + bias ) with Wt row-major [n, ldw]
//    act: 0=none 1=tanh 2=sigmoid.  Used for stab1, stab2, decoder.
// ---------------------------------------------------------------------------
__global__ void bgemm_wmma_kernel(const float* __restrict__ x1,
                                  const float* __restrict__ x2,
                                  int k1, int k2,
                                  const float* __restrict__ Wt, int ldw,
                                  const float* __restrict__ bias,
                                  float* __restrict__ out, int ldout,
                                  int act) {
  int rowbase = blockIdx.x * 16;
  int lane = threadIdx.x;
  int m = lane & 15, kh = lane >> 4;
  int colbase = blockIdx.y * 256 + threadIdx.y * 64;
  int arow = rowbase + m;
  int ktot = k1 + k2;

  v8f acc[4];
#pragma unroll
  for (int ct = 0; ct < 4; ++ct)
#pragma unroll
    for (int i = 0; i < 8; ++i) acc[ct][i] = 0.f;

  for (int kb = 0; kb < ktot; kb += 4) {
    int kk = kb + 2 * kh;
    v2f a;
    if (kk < k1) {
      a.x = x1[(size_t)arow * k1 + kk];
      a.y = x1[(size_t)arow * k1 + kk + 1];
    } else {
      int q = kk - k1;
      a.x = x2[(size_t)arow * k2 + q];
      a.y = x2[(size_t)arow * k2 + q + 1];
    }
#pragma unroll
    for (int ct = 0; ct < 4; ++ct) {
      int n = colbase + ct * 16 + m;
      const float* wp = Wt + (size_t)n * ldw + kk;
      v2f bv; bv.x = wp[0]; bv.y = wp[1];
      acc[ct] = wmma_f32x4(a, bv, acc[ct]);
    }
  }
#pragma unroll
  for (int ct = 0; ct < 4; ++ct) {
    int n = colbase + ct * 16 + m;
    float bb = bias ? bias[n] : 0.f;
#pragma unroll
    for (int i = 0; i < 8; ++i) {
      float v = acc[ct][i] + bb;
      if (act == 1) v = tanhf(v);
      else if (act == 2) v = 1.f / (1.f + expf(-v));
      out[(size_t)(rowbase + i + 8 * kh) * ldout + n] = v;
    }
  }
}

// ---------------------------------------------------------------------------
// 6) h_next = h + gamma[b] * alpha_local * z_star
// ---------------------------------------------------------------------------
__global__ void finish_kernel(const float* __restrict__ h,
                              const float* __restrict__ gamma,
                              const float* __restrict__ al,
                              const float* __restrict__ zs,
                              float* __restrict__ hnext) {
  int idx = blockIdx.x * 256 + threadIdx.x;
  int b = idx >> 10;
  hnext[idx] = h[idx] + gamma[b] * al[idx] * zs[idx];
}

// ---------------------------------------------------------------------------
extern "C" void kernel_launch(void* const* d_in, const int* in_sizes, int n_in,
                              void* d_out, int out_size, void* d_ws,
                              size_t ws_size, hipStream_t stream) {
  const float* h       = (const float*)d_in[0];
  const float* fib_s   = (const float*)d_in[1];
  const float* fib_e   = (const float*)d_in[2];
  const int*   tok     = (const int*)  d_in[3];
  const float* op_emb  = (const float*)d_in[4];
  const float* num_w   = (const float*)d_in[5];
  const float* num_b   = (const float*)d_in[6];
  const float* addr    = (const float*)d_in[7];
  const float* Wm      = (const float*)d_in[8];
  const float* Um      = (const float*)d_in[9];
  const float* Vm      = (const float*)d_in[10];
  const float* fs_w    = (const float*)d_in[11];
  const float* fs_b    = (const float*)d_in[12];
  const float* fe_w    = (const float*)d_in[13];
  const float* fe_b    = (const float*)d_in[14];
  const float* s1w     = (const float*)d_in[15];
  const float* s1b     = (const float*)d_in[16];
  const float* s2w     = (const float*)d_in[17];
  const float* s2b     = (const float*)d_in[18];
  const float* c1w     = (const float*)d_in[19];
  const float* c1b     = (const float*)d_in[20];
  const float* c2w     = (const float*)d_in[21];
  const float* c2b     = (const float*)d_in[22];
  const float* dec_w   = (const float*)d_in[23];
  const float* dec_b   = (const float*)d_in[24];

  float* ws = (float*)d_ws;
  float* femb    = ws + OFF_FEMB;
  float* cbuf    = ws + OFF_C;
  float* zA      = ws + OFF_ZA;
  float* zB      = ws + OFF_ZB;
  float* alocal  = ws + OFF_AL;
  float* stab_h  = ws + OFF_SH;
  float* gamma   = ws + OFF_GAMMA;
  int*   expert  = (int*)(ws + OFF_EXPERT);
  int*   texp    = (int*)(ws + OFF_TEXP);
  int*   trows   = (int*)(ws + OFF_TROWS);
  int*   ntiles  = (int*)(ws + OFF_NT);

  float* out        = (float*)d_out;
  float* out_hnext  = out;                              // [256*1024]
  float* out_logits = out + (size_t)B_SZ * D_SZ;        // [256*32000]
  float* out_pi     = out_logits + (size_t)B_SZ * VOC_SZ; // [256*7]

  // 1) routing / embeddings / gamma
  prep_kernel<<<B_SZ, 256, 0, stream>>>(h, fib_s, fib_e, tok, op_emb, num_w,
                                        num_b, addr, fs_w, fs_b, fe_w, fe_b,
                                        c1w, c1b, c2w, c2b, femb, gamma,
                                        expert, out_pi);
  // 2) bucket rows by expert into 16-row WMMA tiles
  group_kernel<<<1, 32, 0, stream>>>(expert, texp, trows, ntiles);

  dim3 tgrid(MAXT, D_SZ / 256), tblk(32, 4);
  // 3) c = h@U + f@V ; z1 = tanh(c)
  deq_const_kernel<<<tgrid, tblk, 0, stream>>>(h, femb, Um, Vm, texp, trows,
                                               ntiles, cbuf, zA);
  // 4) 39 remaining fixed-point iterations (40 evals total, as in reference)
  float* zin = zA; float* zout = zB;
  for (int it = 0; it < 39; ++it) {
    deq_iter_kernel<<<tgrid, tblk, 0, stream>>>(zin, zout, cbuf, Wm, texp,
                                                trows, ntiles);
    float* t = zin; zin = zout; zout = t;
  }
  float* z_star = zin;  // result of last iteration

  // 5) stab_h = tanh([h|f] @ s1w^T + s1b); alpha_local = sigmoid(stab_h @ s2w^T + s2b)
  bgemm_wmma_kernel<<<dim3(B_SZ / 16, D_SZ / 256), tblk, 0, stream>>>(
      h, femb, D_SZ, D_SZ, s1w, 2 * D_SZ, s1b, stab_h, D_SZ, 1);
  bgemm_wmma_kernel<<<dim3(B_SZ / 16, D_SZ / 256), tblk, 0, stream>>>(
      stab_h, stab_h, D_SZ, 0, s2w, D_SZ, s2b, alocal, D_SZ, 2);

  // 6) h_next = h + gamma * alpha_local * z_star
  finish_kernel<<<(B_SZ * D_SZ) / 256, 256, 0, stream>>>(h, gamma, alocal,
                                                         z_star, out_hnext);
  // 7) logits = h_next @ dec_w^T + dec_b
  bgemm_wmma_kernel<<<dim3(B_SZ / 16, VOC_SZ / 256), tblk, 0, stream>>>(
      out_hnext, out_hnext, D_SZ, 0, dec_w, D_SZ, dec_b, out_logits, VOC_SZ, 0);
}